// MidGCN_953482740188
// MI455X (gfx1250) — compile-verified
//
#include <hip/hip_runtime.h>
#include <hip/hip_bf16.h>
#include <math.h>
#include <stdint.h>

#define NNODES 8192
#define NFEAT  512
#define NHID   256
#define NCLASS 2

#define BM 128
#define BN 64
#define BK 32
#define AS_STRIDE 36   // 144B rows: 16B aligned, 36*m distinct mod 64 -> conflict-free b64 reads
#define BS_STRIDE 80   // 320B rows: 16B aligned, rows +2 land 32 banks away

typedef float v2f __attribute__((ext_vector_type(2)));
typedef float v8f __attribute__((ext_vector_type(8)));

// -------------------- gfx1250 async global->LDS copy (ASYNCcnt) --------------------
__device__ __forceinline__ void async_copy_b128(uint32_t lds_off, const float* gptr) {
    asm volatile("global_load_async_to_lds_b128 %0, %1, off"
                 :: "v"(lds_off), "v"((uint64_t)(uintptr_t)gptr)
                 : "memory");
}

// -------------------- degree vectors: d = rsqrt(sum), inf -> 0 --------------------
__global__ __launch_bounds__(256)
void colsum_rsqrt_kernel(const float* __restrict__ adj, float* __restrict__ dcol) {
    int j = blockIdx.x * 256 + threadIdx.x;
    float s = 0.f;
    for (int i = 0; i < NNODES; ++i) s += adj[(size_t)i * NNODES + j];
    dcol[j] = (s > 0.f) ? rsqrtf(s) : 0.f;
}

__global__ __launch_bounds__(256)
void rowsum_rsqrt_kernel(const float* __restrict__ adj, float* __restrict__ drow) {
    __shared__ float red[256];
    int i = blockIdx.x;
    float s = 0.f;
    const float* arow = adj + (size_t)i * NNODES;
    for (int j = threadIdx.x; j < NNODES; j += 256) s += arow[j];
    red[threadIdx.x] = s;
    __syncthreads();
    for (int o = 128; o > 0; o >>= 1) {
        if (threadIdx.x < o) red[threadIdx.x] += red[threadIdx.x + o];
        __syncthreads();
    }
    if (threadIdx.x == 0) {
        float t = red[0];
        drow[i] = (t > 0.f) ? rsqrtf(t) : 0.f;
    }
}

// -------------------- Bout[k,:] = dcol[k] * Bin[k,:]  (N == NHID == 256) --------------------
__global__ __launch_bounds__(256)
void scale_rows_kernel(const float* __restrict__ Bin, const float* __restrict__ dcol,
                       float* __restrict__ Bout) {
    size_t idx = (size_t)blockIdx.x * 256 + threadIdx.x;
    Bout[idx] = Bin[idx] * dcol[idx >> 8];
}

// -------------------- fp32 WMMA GEMM: C = drow ⊙ (A @ B) --------------------
// A: NNODES x K row-major, B: K x Ncols row-major (pre-scaled by dcol if needed).
// 128x64 block tile, BK=32, double-buffered LDS fed by async B128 copies.
// 8 waves: wave w -> rows (w&3)*32, cols (w>>2)*32; 2x2 V_WMMA_F32_16X16X4_F32 accs.
__global__ __launch_bounds__(256)
void wmma_gemm_async(const float* __restrict__ A, const float* __restrict__ B,
                     float* __restrict__ C, int K, int Ncols,
                     const float* __restrict__ drow)
{
    __shared__ float As[2][BM][AS_STRIDE];
    __shared__ float Bs[2][BK][BS_STRIDE];

    const int tid  = threadIdx.x;
    const int wave = tid >> 5;
    const int lane = tid & 31;
    const int h    = lane >> 4;
    const int lm   = lane & 15;

    const int colBase = blockIdx.x * BN;   // col tile fast -> siblings share adj strip in L2
    const int rowBase = blockIdx.y * BM;
    const int wr = (wave & 3) * 32;
    const int wc = (wave >> 2) * 32;

    const uint32_t asBase = (uint32_t)(size_t)(&As[0][0][0]);
    const uint32_t bsBase = (uint32_t)(size_t)(&Bs[0][0][0]);
    const uint32_t asBufBytes = BM * AS_STRIDE * 4u;
    const uint32_t bsBufBytes = BK * BS_STRIDE * 4u;

    v8f acc[2][2];
#pragma unroll
    for (int i = 0; i < 2; ++i)
#pragma unroll
        for (int j = 0; j < 2; ++j)
#pragma unroll
            for (int e = 0; e < 8; ++e) acc[i][j][e] = 0.f;

    // 6 async B128 per thread per tile: A = 128x32 fl (1024 16B chunks), B = 32x64 fl (512 chunks)
    auto issue_tile = [&](int buf, int kk) {
#pragma unroll
        for (int i = 0; i < 4; ++i) {
            int c   = tid + 256 * i;
            int r   = c >> 3;
            int col = (c & 7) * 4;
            uint32_t dst = asBase + (uint32_t)buf * asBufBytes + (uint32_t)(r * AS_STRIDE + col) * 4u;
            async_copy_b128(dst, &A[(size_t)(rowBase + r) * K + kk + col]);
        }
#pragma unroll
        for (int i = 0; i < 2; ++i) {
            int c   = tid + 256 * i;
            int kr  = c >> 4;
            int col = (c & 15) * 4;
            uint32_t dst = bsBase + (uint32_t)buf * bsBufBytes + (uint32_t)(kr * BS_STRIDE + col) * 4u;
            async_copy_b128(dst, &B[(size_t)(kk + kr) * Ncols + colBase + col]);
        }
    };

    const int nIter = K / BK;
    issue_tile(0, 0);

    for (int it = 0; it < nIter; ++it) {
        const int cur = it & 1;
        if (it + 1 < nIter) {
            issue_tile(cur ^ 1, (it + 1) * BK);           // prefetch next buffer
            asm volatile("s_wait_asynccnt 6" ::: "memory"); // oldest 6 (cur buffer) done, 6 in flight
        } else {
            asm volatile("s_wait_asynccnt 0" ::: "memory");
        }
        __syncthreads();   // all waves' cur-buffer data landed

#pragma unroll
        for (int k2 = 0; k2 < BK; k2 += 4) {
            v2f afrag[2], bfrag[2];
            // fp32 16x4 A layout: lane h*16+m holds A[m][2h], A[m][2h+1]
#pragma unroll
            for (int rs = 0; rs < 2; ++rs) {
                afrag[rs][0] = As[cur][wr + rs * 16 + lm][k2 + 2 * h + 0];
                afrag[rs][1] = As[cur][wr + rs * 16 + lm][k2 + 2 * h + 1];
            }
            // fp32 4x16 B layout (mirror): lane h*16+n holds B[2h][n], B[2h+1][n]
#pragma unroll
            for (int cs = 0; cs < 2; ++cs) {
                bfrag[cs][0] = Bs[cur][k2 + 2 * h + 0][wc + cs * 16 + lm];
                bfrag[cs][1] = Bs[cur][k2 + 2 * h + 1][wc + cs * 16 + lm];
            }
#pragma unroll
            for (int rs = 0; rs < 2; ++rs)
#pragma unroll
                for (int cs = 0; cs < 2; ++cs)
                    acc[rs][cs] = __builtin_amdgcn_wmma_f32_16x16x4_f32(
                        false, afrag[rs], false, bfrag[cs],
                        (short)0, acc[rs][cs], false, false);
        }
        __syncthreads();   // everyone done reading cur before next issue overwrites it
    }

    // C/D layout: VGPR v -> row (v + 8h), col lm within the 16x16 tile
#pragma unroll
    for (int rs = 0; rs < 2; ++rs)
#pragma unroll
        for (int cs = 0; cs < 2; ++cs)
#pragma unroll
            for (int v = 0; v < 8; ++v) {
                int row = rowBase + wr + rs * 16 + h * 8 + v;
                int col = colBase + wc + cs * 16 + lm;
                float sc = drow ? drow[row] : 1.0f;
                C[(size_t)row * Ncols + col] = acc[rs][cs][v] * sc;
            }
}

// -------------------- h = relu(0.5*XW1 - 0.5*T1 - T2), in-place on XW1 --------------------
__global__ __launch_bounds__(256)
void combine_relu_kernel(float* __restrict__ xw1_h, const float* __restrict__ t1,
                         const float* __restrict__ t2) {
    size_t idx = (size_t)blockIdx.x * 256 + threadIdx.x;
    float v = 0.5f * xw1_h[idx] - 0.5f * t1[idx] - t2[idx];
    xw1_h[idx] = fmaxf(v, 0.f);
}

// -------------------- U = h @ W2  (one wave32 per row, coalesced) --------------------
__global__ __launch_bounds__(256)
void hw2_kernel(const float* __restrict__ h, const float* __restrict__ W2,
                float* __restrict__ U) {
    int wave = threadIdx.x >> 5;
    int lane = threadIdx.x & 31;
    int i = blockIdx.x * 8 + wave;
    const float* hr = h + (size_t)i * NHID;
    float u0 = 0.f, u1 = 0.f;
    for (int k = lane; k < NHID; k += 32) {
        float hv = hr[k];
        u0 = fmaf(hv, W2[2 * k + 0], u0);
        u1 = fmaf(hv, W2[2 * k + 1], u1);
    }
#pragma unroll
    for (int m = 16; m >= 1; m >>= 1) {
        u0 += __shfl_xor(u0, m, 32);
        u1 += __shfl_xor(u1, m, 32);
    }
    if (lane == 0) {
        U[2 * i + 0] = u0;
        U[2 * i + 1] = u1;
    }
}

// -------------------- V = drow ⊙ (adj @ (dcol ⊙ Uin)), N==2 (bandwidth-bound) --------------------
__global__ __launch_bounds__(256)
void dad2_kernel(const float* __restrict__ adj, const float* __restrict__ Uin,
                 float* __restrict__ Vout, const float* __restrict__ drow,
                 const float* __restrict__ dcol) {
    __shared__ float r0[256], r1[256];
    int i = blockIdx.x;
    const float* arow = adj + (size_t)i * NNODES;
    float s0 = 0.f, s1 = 0.f;
    for (int j = threadIdx.x; j < NNODES; j += 256) {
        float a = arow[j] * dcol[j];
        s0 = fmaf(a, Uin[2 * j + 0], s0);
        s1 = fmaf(a, Uin[2 * j + 1], s1);
    }
    r0[threadIdx.x] = s0;
    r1[threadIdx.x] = s1;
    __syncthreads();
    for (int o = 128; o > 0; o >>= 1) {
        if (threadIdx.x < o) {
            r0[threadIdx.x] += r0[threadIdx.x + o];
            r1[threadIdx.x] += r1[threadIdx.x + o];
        }
        __syncthreads();
    }
    if (threadIdx.x == 0) {
        float sc = drow[i];
        Vout[2 * i + 0] = r0[0] * sc;
        Vout[2 * i + 1] = r1[0] * sc;
    }
}

// -------------------- out = log_softmax(0.5U - 0.5V1 - V2 + b2) --------------------
__global__ __launch_bounds__(256)
void final_kernel(const float* __restrict__ U, const float* __restrict__ V1,
                  const float* __restrict__ V2, const float* __restrict__ b2,
                  float* __restrict__ out) {
    int i = blockIdx.x * 256 + threadIdx.x;
    float z0 = 0.5f * U[2 * i + 0] - 0.5f * V1[2 * i + 0] - V2[2 * i + 0] + b2[0];
    float z1 = 0.5f * U[2 * i + 1] - 0.5f * V1[2 * i + 1] - V2[2 * i + 1] + b2[1];
    float m = fmaxf(z0, z1);
    float lse = m + logf(expf(z0 - m) + expf(z1 - m));
    out[2 * i + 0] = z0 - lse;
    out[2 * i + 1] = z1 - lse;
}

extern "C" void kernel_launch(void* const* d_in, const int* in_sizes, int n_in,
                              void* d_out, int out_size, void* d_ws, size_t ws_size,
                              hipStream_t stream) {
    const float* x   = (const float*)d_in[0];  // 8192 x 512
    const float* adj = (const float*)d_in[1];  // 8192 x 8192
    const float* W1  = (const float*)d_in[2];  // 512 x 256
    const float* W2  = (const float*)d_in[3];  // 256 x 2
    const float* b2  = (const float*)d_in[4];  // 2
    float* out = (float*)d_out;                // 8192 x 2

    float* ws   = (float*)d_ws;
    float* drow = ws;                               // 8192
    float* dcol = drow + NNODES;                    // 8192
    float* XW1  = dcol + NNODES;                    // 8192*256 (becomes h in-place)
    float* T1   = XW1 + (size_t)NNODES * NHID;      // 8192*256
    float* T2   = T1  + (size_t)NNODES * NHID;      // 8192*256
    float* Bsc  = T2  + (size_t)NNODES * NHID;      // 8192*256 (dcol-scaled B scratch)
    float* U    = Bsc + (size_t)NNODES * NHID;      // 8192*2
    float* V1   = U   + NNODES * NCLASS;            // 8192*2
    float* V2   = V1  + NNODES * NCLASS;            // 8192*2  (~34 MB total)

    // degrees
    colsum_rsqrt_kernel<<<NNODES / 256, 256, 0, stream>>>(adj, dcol);
    rowsum_rsqrt_kernel<<<NNODES, 256, 0, stream>>>(adj, drow);

    dim3 g(NHID / BN, NNODES / BM);   // (4, 64): col tiles fast for L2 adj-strip reuse

    // XW1 = x @ W1 (no scaling)
    wmma_gemm_async<<<g, 256, 0, stream>>>(x, W1, XW1, NFEAT, NHID, nullptr);

    // T1 = drow ⊙ (adj @ (dcol ⊙ XW1))
    scale_rows_kernel<<<(NNODES * NHID) / 256, 256, 0, stream>>>(XW1, dcol, Bsc);
    wmma_gemm_async<<<g, 256, 0, stream>>>(adj, Bsc, T1, NNODES, NHID, drow);

    // T2 = drow ⊙ (adj @ (dcol ⊙ T1))
    scale_rows_kernel<<<(NNODES * NHID) / 256, 256, 0, stream>>>(T1, dcol, Bsc);
    wmma_gemm_async<<<g, 256, 0, stream>>>(adj, Bsc, T2, NNODES, NHID, drow);

    // h = relu(0.5*XW1 - 0.5*T1 - T2)   (in-place over XW1)
    combine_relu_kernel<<<(NNODES * NHID) / 256, 256, 0, stream>>>(XW1, T1, T2);

    // layer 2
    hw2_kernel<<<NNODES / 8, 256, 0, stream>>>(XW1, W2, U);
    dad2_kernel<<<NNODES, 256, 0, stream>>>(adj, U, V1, drow, dcol);
    dad2_kernel<<<NNODES, 256, 0, stream>>>(adj, V1, V2, drow, dcol);
    final_kernel<<<NNODES / 256, 256, 0, stream>>>(U, V1, V2, b2, out);
}